// Temporal_816043786817
// MI455X (gfx1250) — compile-verified
//
#include <hip/hip_runtime.h>
#include <cstdint>

// Problem dims (from reference): x is [8, 16, 3, 224, 224] f32,
// out is [8, 16, 224, 224, 3] f32 (permute (0,1,4,3,2)).
#define B   8
#define T   16
#define C   3
#define H   224
#define W   224
#define HW  (H * W)        // 50176
#define CHW (C * HW)       // 150528 elements per (b, t)
#define P4  (CHW * T / 4)  // 602112 float4 per batch

// Tile for the batch-0 recurrence+transpose path
#define TH_   8
#define TW_   32
#define NTW   (W / TW_)    // 7
#define NTH   (H / TH_)    // 28
#define NTILE (NTW * NTH)  // 196
#define TRPITCH 28         // 24 payload floats per w, padded: 28*4=112 B (16B aligned)

#define FILL_BLOCKS ((7 * P4) / 256)       // 16464
#define TOTAL_BLOCKS (NTILE + FILL_BLOCKS) // 16660

__device__ __forceinline__ unsigned lds_byte_off(const void* p) {
  // Generic (flat) pointers to LDS carry the LDS byte offset in the low 32 bits.
  return (unsigned)(uintptr_t)p;
}

// ---------------------------------------------------------------------------
// Single fused kernel:
//   blocks [0, NTILE):            batch 0 — temporal recurrence + transpose,
//                                 async global->LDS double-buffered staging
//                                 (CDNA5 ASYNCcnt path), LDS transpose,
//                                 coalesced b128 stores.
//   blocks [NTILE, TOTAL_BLOCKS): batches 1..6 zeros, batch 7 broadcast scalar
//                                 recurrence, pure b128 store stream.
// Fusing lets the 16464 fill blocks saturate HBM while the 196 recurrence
// blocks work through their serial 16-frame chains -> ~one-pass runtime.
// ---------------------------------------------------------------------------
__global__ __launch_bounds__(256) void temporal_fused(const float* __restrict__ x,
                                                      float* __restrict__ out) {
  const int tid = threadIdx.x;
  const int bid = blockIdx.x;

  if (bid >= NTILE) {
    // ---------------- Fill path: batches 1..7 ----------------
    const int idx   = (bid - NTILE) * 256 + tid;   // 0 .. 7*P4-1 (exact)
    const int b_rel = idx / P4;                    // 0..6 => batches 1..7

    float4 v = make_float4(0.f, 0.f, 0.f, 0.f);
    if (b_rel == 6) {                              // batch b-1
      const int rem = idx - 6 * P4;
      const int tt  = rem / (CHW / 4);             // frame index 0..15
      float z = 0.f;
      for (int i = 0; i < tt; ++i)
        z = __fsub_rn(3.0f, __fmul_rn(0.4f, z));   // exact f32 scan replay
      v = make_float4(z, z, z, z);
    }
    ((float4*)out)[P4 + idx] = v;                  // offset past batch 0
    return;
  }

  // ---------------- Batch-0 recurrence + (c,h,w)->(w,h,c) transpose ----------
  __shared__ float stage[2][C * 256];      // [buf][c*256 + tid] staged x values
  __shared__ float tr[TW_ * TRPITCH];      // transposed tile: [w_local][hloc*3 + c]

  const int tw     = bid % NTW;
  const int th     = bid / NTW;
  const int w0     = tw * TW_;
  const int lane_w = tid & (TW_ - 1);      // 0..31 -> coalesced along w
  const int hloc   = tid >> 5;             // 0..7
  const int pix    = (th * TH_ + hloc) * W + (w0 + lane_w);

  // Pre-issue async loads for frame 0 into buffer 0.
#pragma unroll
  for (int c = 0; c < C; ++c) {
    unsigned go = (unsigned)((c * HW + pix) * 4);
    unsigned lo = lds_byte_off(&stage[0][c * 256 + tid]);
    asm volatile("global_load_async_to_lds_b32 %0, %1, %2"
                 :: "v"(lo), "v"(go), "s"(x) : "memory");
  }

  float carry[C] = {0.f, 0.f, 0.f};

  for (int f = 0; f < T; ++f) {
    const int buf = f & 1;

    if (f + 1 < T) {
      // Issue next frame's loads into the other buffer, then wait for the
      // oldest 3 (this frame's) to land; async loads complete in order.
#pragma unroll
      for (int c = 0; c < C; ++c) {
        unsigned go = (unsigned)(((f + 1) * CHW + c * HW + pix) * 4);
        unsigned lo = lds_byte_off(&stage[buf ^ 1][c * 256 + tid]);
        asm volatile("global_load_async_to_lds_b32 %0, %1, %2"
                     :: "v"(lo), "v"(go), "s"(x) : "memory");
      }
      asm volatile("s_wait_asynccnt 0x3" ::: "memory");
    } else {
      asm volatile("s_wait_asynccnt 0x0" ::: "memory");
    }

    // Recurrence (matches JAX eval order: (x - 0.4*c) + 3, no FMA contraction)
    // and scatter into the transposed LDS tile.
#pragma unroll
    for (int c = 0; c < C; ++c) {
      const float xv = stage[buf][c * 256 + tid];
      float y;
      if (f == 0) {
        y = xv;                                  // y[0] = x[0]
      } else {
        y = __fadd_rn(__fsub_rn(xv, __fmul_rn(0.4f, carry[c])), 3.0f);
      }
      carry[c] = y;
      tr[lane_w * TRPITCH + hloc * 3 + c] = y;
    }
    __syncthreads();

    // Coalesced b128 stores: per w, 24 contiguous floats (8h x 3c) = 6 float4.
    if (tid < TW_ * 6) {
      const int wl = tid / 6;
      const int q  = tid % 6;
      const float4 v = *(const float4*)&tr[wl * TRPITCH + q * 4];
      const int o4 = f * (CHW / 4) + (w0 + wl) * (H * C / 4) + th * (TH_ * C / 4) + q;
      ((float4*)out)[o4] = v;
    }
    __syncthreads();
  }
}

// ---------------------------------------------------------------------------
extern "C" void kernel_launch(void* const* d_in, const int* in_sizes, int n_in,
                              void* d_out, int out_size, void* d_ws, size_t ws_size,
                              hipStream_t stream) {
  (void)in_sizes; (void)n_in; (void)out_size; (void)d_ws; (void)ws_size;
  const float* x = (const float*)d_in[0];
  float* out = (float*)d_out;

  temporal_fused<<<TOTAL_BLOCKS, 256, 0, stream>>>(x, out);
}